// VertexConv_39084202394049
// MI455X (gfx1250) — compile-verified
//
#include <hip/hip_runtime.h>
#include <hip/hip_bf16.h>
#include <math.h>

typedef __attribute__((ext_vector_type(2))) float v2f;
typedef __attribute__((ext_vector_type(8))) float v8f;

#define KV 32       // vertices per edge (== wave32)
#define DIMV 128    // feature dim
#define XPITCH 132  // LDS row pitch (pad to kill 64-bank stride conflicts; 132*4B is 16B aligned)
#define WPITCH 132

__global__ __launch_bounds__(128) void vertexconv_kernel(
    const float* __restrict__ feats,
    const int*   __restrict__ edges,
    const float* __restrict__ wq, const float* __restrict__ bq,
    const float* __restrict__ wk, const float* __restrict__ bk,
    const float* __restrict__ wv, const float* __restrict__ bv,
    float* __restrict__ out, int n_edges)
{
    __shared__ float Xs[KV * XPITCH];   // gathered 32x128 tile (padded)
    __shared__ float Ws[4 * WPITCH];    // [wq | wk | wv | zeros], bank-staggered
    __shared__ float qs[KV], ks[KV], vs[KV], dgate[KV];

    const int t = threadIdx.x;
    const int e = blockIdx.x;
    if (e >= n_edges) return;

    // ---- stage projection weights (+ explicit zero column for unused WMMA-B lanes)
    if (t < DIMV) {
        Ws[0 * WPITCH + t] = wq[t];
        Ws[1 * WPITCH + t] = wk[t];
        Ws[2 * WPITCH + t] = wv[t];
        Ws[3 * WPITCH + t] = 0.0f;
    }

    // ---- gather X = feats[edges[e]] : 32 rows x 128 cols -> LDS (float4, 16KB/edge)
    {
        const int j     = t >> 2;          // vertex 0..31 (4 threads per row)
        const int cbase = (t & 3) * 32;    // 32-col slab per thread
        const long long row = (long long)edges[e * KV + j];
        const float4* src = (const float4*)(feats + row * (long long)DIMV + cbase);
        float4* dst = (float4*)(&Xs[j * XPITCH + cbase]);
#pragma unroll
        for (int it = 0; it < 8; ++it) dst[it] = src[it];
    }
    __syncthreads();

    // ---- QKV projection: P = X[32x128] @ [wq wk wv 0..0][128x16]
    //      via v_wmma_f32_16x16x4_f32, wave0 -> rows 0..15, wave1 -> rows 16..31
    const int wave = t >> 5;
    if (wave < 2) {
        const int lane = t & 31;
        const int half = lane >> 4;            // 0: K+0/K+1, 1: K+2/K+3 (A & B sub-K)
        const int n    = lane & 15;            // A row within tile / B(=D) column
        const int rowA = wave * 16 + n;
        const int koff = 2 * half;
        const float* wp = &Ws[(n < 3 ? n : 3) * WPITCH];  // branch-free: EXEC stays all-1s
        v8f c = {};
#pragma unroll
        for (int kk = 0; kk < DIMV; kk += 4) {
            const float* xp = &Xs[rowA * XPITCH + kk + koff];
            v2f a, b;
            a.x = xp[0];              a.y = xp[1];
            b.x = wp[kk + koff];      b.y = wp[kk + koff + 1];
            // (neg_a, A, neg_b, B, c_mod, C, reuse_a, reuse_b)
            c = __builtin_amdgcn_wmma_f32_16x16x4_f32(false, a, false, b,
                                                      (short)0, c, false, false);
        }
        // D layout: column n lives in lanes n (rows 0..7 in c[0..7]) and n+16 (rows 8..15)
        if (n < 3) {
            float* dstp = (n == 0) ? qs : (n == 1) ? ks : vs;
            const int base = wave * 16 + half * 8;
#pragma unroll
            for (int r = 0; r < 8; ++r) dstp[base + r] = c[r];
        }
    }
    __syncthreads();

    // ---- masked softmax attention + tanh gate: lane j of wave0 handles vertex j
    //      (branchless diagonal mask: cndmask to -inf, so exp() term is exactly 0)
    if (t < KV) {
        const float bqv = bq[0], bkv = bk[0], bvv = bv[0];
        const float qj   = qs[t] + bqv;
        const float NINF = -INFINITY;
        float maxl = NINF;
#pragma unroll
        for (int m = 0; m < KV; ++m) {
            float l = qj * (ks[m] + bkv);
            l = (m == t) ? NINF : l;
            maxl = fmaxf(maxl, l);
        }
        float se = 0.f, sv = 0.f;
#pragma unroll
        for (int m = 0; m < KV; ++m) {
            float l = qj * (ks[m] + bkv);
            l = (m == t) ? NINF : l;
            const float ex = __expf(l - maxl);   // 0 for the masked diagonal
            se += ex;
            sv += ex * (vs[m] + bvv);
        }
        dgate[t] = tanhf(sv / se);
    }
    __syncthreads();

    // ---- out[e] = sum_j d_j * X[j,:]  (thread t owns dim t; conflict-free column walk)
    {
        float acc = 0.f;
#pragma unroll
        for (int j = 0; j < KV; ++j) acc += dgate[j] * Xs[j * XPITCH + t];
        out[(long long)e * DIMV + t] = acc;
    }
}

extern "C" void kernel_launch(void* const* d_in, const int* in_sizes, int n_in,
                              void* d_out, int out_size, void* d_ws, size_t ws_size,
                              hipStream_t stream) {
    const float* feats = (const float*)d_in[0];
    const int*   edges = (const int*)d_in[1];
    const float* wq    = (const float*)d_in[2];
    const float* bq    = (const float*)d_in[3];
    const float* wk    = (const float*)d_in[4];
    const float* bk    = (const float*)d_in[5];
    const float* wv    = (const float*)d_in[6];
    const float* bv    = (const float*)d_in[7];
    float* out = (float*)d_out;

    const int n_edges = in_sizes[1] / KV;   // 50000
    dim3 grid(n_edges), block(128);
    vertexconv_kernel<<<grid, block, 0, stream>>>(feats, edges, wq, bq, wk, bk,
                                                  wv, bv, out, n_edges);
}